// GcnNet_72035191488553
// MI455X (gfx1250) — compile-verified
//
#include <hip/hip_runtime.h>

typedef float v2f __attribute__((ext_vector_type(2)));
typedef float v8f __attribute__((ext_vector_type(8)));

#define N_POOL   100000
#define N_POINTS 400000
#define N_EDGES  3200000

// ---- CDNA5 async global->LDS staging (ASYNCcnt path), guarded so compile is safe
#if defined(__AMDGCN__) && __has_builtin(__builtin_amdgcn_global_load_async_to_lds_b128) && __has_builtin(__builtin_amdgcn_s_wait_asynccnt)
#define USE_ASYNC_LDS 1
#else
#define USE_ASYNC_LDS 0
#endif

typedef int v4i __attribute__((vector_size(16)));
typedef __attribute__((address_space(1))) void as1_void;
typedef __attribute__((address_space(3))) void as3_void;
typedef __attribute__((address_space(1))) v4i as1_v4i;
typedef __attribute__((address_space(3))) v4i as3_v4i;

__device__ __forceinline__ void stage16(const float* __restrict__ g, float* l) {
#if USE_ASYNC_LDS
  __builtin_amdgcn_global_load_async_to_lds_b128(
      (as1_v4i*)(as1_void*)(void*)g,
      (as3_v4i*)(as3_void*)(void*)l, 0, 0);
#else
  *(float4*)l = *(const float4*)g;
#endif
}

__device__ __forceinline__ void stage_wait() {
#if USE_ASYNC_LDS
  __builtin_amdgcn_s_wait_asynccnt(0);
#endif
  __syncthreads();
}

// ---------------------------------------------------------------- utilities
__global__ void k_zero(float* __restrict__ p, int n) {
  int i = blockIdx.x * 256 + threadIdx.x;
  if (i < n) p[i] = 0.f;
}

// ------------------------------------------------- gemm1: feature(100k,4) @ w1(4,128) -> xw1
__global__ __launch_bounds__(256) void k_gemm1(const float* __restrict__ feat,
                                               const float* __restrict__ w1,
                                               float* __restrict__ xw1) {
  int lane = threadIdx.x & 31, wv = threadIdx.x >> 5;
  int lo = lane & 15, hi = lane >> 4;
  int m0 = blockIdx.x * 16;
  int n0 = wv * 16;
  v2f a = *(const v2f*)(feat + (m0 + lo) * 4 + 2 * hi);   // K = 2*hi + {0,1}
  v2f b;
  b.x = w1[(2 * hi + 0) * 128 + n0 + lo];
  b.y = w1[(2 * hi + 1) * 128 + n0 + lo];
  v8f c = {};
  c = __builtin_amdgcn_wmma_f32_16x16x4_f32(false, a, false, b, (short)0, c, false, false);
#pragma unroll
  for (int v = 0; v < 8; ++v)
    xw1[(m0 + v + 8 * hi) * 128 + n0 + lo] = c[v];
}

// ------------------------------------------------- spmm1: agg1[row] += val * xw1[col], C=128
__global__ void k_spmm1(const int* __restrict__ rows, const int* __restrict__ cols,
                        const float* __restrict__ vals, const float* __restrict__ x,
                        float* __restrict__ agg) {
  int t = blockIdx.x * 256 + threadIdx.x;
  int e = t >> 7, c = t & 127;
  int r = rows[e], s = cols[e];
  unsafeAtomicAdd(&agg[r * 128 + c], vals[e] * x[s * 128 + c]);
}

// ------------------------------------------------- h1 = [relu(agg1+b1) | feature]  (100k,132)
__global__ void k_h1(const float* __restrict__ agg, const float* __restrict__ b1,
                     const float* __restrict__ feat, float* __restrict__ h1) {
  int i = blockIdx.x * 256 + threadIdx.x;
  if (i >= N_POOL * 132) return;
  int r = i / 132, c = i - r * 132;
  float v;
  if (c < 128) { v = agg[r * 128 + c] + b1[c]; v = v > 0.f ? v : 0.f; }
  else          v = feat[r * 4 + (c - 128)];
  h1[i] = v;
}

// ------------------------------------------------- gemm2: h1(100k,132) @ w2(132,256) -> xw2
// 32 rows / block (2 M-subtiles per wave -> B fragment reused across 2 WMMAs)
__global__ __launch_bounds__(256) void k_gemm2(const float* __restrict__ A,
                                               const float* __restrict__ w2,
                                               float* __restrict__ out) {
  __shared__ __align__(16) float sA[32 * 132];
  int tid = threadIdx.x;
  int m0 = blockIdx.x * 32;
  for (int idx = tid; idx < 32 * 33; idx += 256) {         // 33 x 16B chunks per row
    int r = idx / 33, c = idx - r * 33;
    stage16(A + (m0 + r) * 132 + c * 4, sA + r * 132 + c * 4);
  }
  stage_wait();
  int lane = tid & 31, wv = tid >> 5;
  int lo = lane & 15, hi = lane >> 4;
#pragma unroll
  for (int nt = 0; nt < 2; ++nt) {
    int n0 = (wv * 2 + nt) * 16;
    v8f c0 = {}, c1 = {};
    for (int k0 = 0; k0 < 132; k0 += 4) {
      v2f b;
      b.x = w2[(k0 + 2 * hi + 0) * 256 + n0 + lo];
      b.y = w2[(k0 + 2 * hi + 1) * 256 + n0 + lo];
      v2f a0 = *(const v2f*)(sA + lo * 132 + k0 + 2 * hi);
      v2f a1 = *(const v2f*)(sA + (16 + lo) * 132 + k0 + 2 * hi);
      c0 = __builtin_amdgcn_wmma_f32_16x16x4_f32(false, a0, false, b, (short)0, c0, false, false);
      c1 = __builtin_amdgcn_wmma_f32_16x16x4_f32(false, a1, false, b, (short)0, c1, false, false);
    }
#pragma unroll
    for (int v = 0; v < 8; ++v) {
      out[(m0 + v + 8 * hi) * 256 + n0 + lo]      = c0[v];
      out[(m0 + 16 + v + 8 * hi) * 256 + n0 + lo] = c1[v];
    }
  }
}

// ------------------------------------------------- spmm2: C=256
__global__ void k_spmm2(const int* __restrict__ rows, const int* __restrict__ cols,
                        const float* __restrict__ vals, const float* __restrict__ x,
                        float* __restrict__ agg) {
  int e = blockIdx.x;
  int c = threadIdx.x;                  // 256 cols
  int r = rows[e], s = cols[e];
  unsafeAtomicAdd(&agg[r * 256 + c], vals[e] * x[s * 256 + c]);
}

// ------------------------------------------------- h2a = relu(agg2 + b2) in place
__global__ void k_h2a(float* __restrict__ agg, const float* __restrict__ b2) {
  int i = blockIdx.x * 256 + threadIdx.x;   // exactly 25.6M
  int c = i & 255;
  float v = agg[i] + b2[c];
  agg[i] = v > 0.f ? v : 0.f;
}

// --------------- gemm3: h2cat(400k,260) @ w3(260,512), bias+relu -> h3 and h3_cat[:,0:512]
// h2cat row i = [bing_vals[i]*h2a[bing_cols[i],:256] | tensor_x3[i,:4]].
// Stage RAW h2a rows (async b128, padded stride 260); apply the per-row scale on the
// A fragment at read time ((w*x)*b == same fp32 FMA); the x3 tail is the final K-step
// with fragments loaded directly from global. 32 rows/block -> B reuse across 2 WMMAs.
__global__ __launch_bounds__(256) void k_gemm3(const float* __restrict__ h2a,
                                               const int*   __restrict__ bcols,
                                               const float* __restrict__ bvals,
                                               const float* __restrict__ x3,
                                               const float* __restrict__ w3,
                                               const float* __restrict__ b3,
                                               float* __restrict__ h3,
                                               float* __restrict__ h3cat) {
  __shared__ __align__(16) float sA[32 * 260];   // rows padded 256 -> 260 (bank-conflict free)
  int tid = threadIdx.x;
  int m0 = blockIdx.x * 32;
  for (int idx = tid; idx < 32 * 64; idx += 256) {          // 64 x 16B chunks per row
    int r = idx >> 6, c4 = idx & 63;
    stage16(h2a + (long long)bcols[m0 + r] * 256 + c4 * 4, sA + r * 260 + c4 * 4);
  }
  if (tid < 128) {   // fused concat tail: h3_cat[:,512:516] = tensor_x3
    int r = tid >> 2, c = tid & 3;
    h3cat[(long long)(m0 + r) * 516 + 512 + c] = x3[(m0 + r) * 4 + c];
  }
  stage_wait();
  int lane = tid & 31, wv = tid >> 5;
  int lo = lane & 15, hi = lane >> 4;
  float s0 = bvals[m0 + lo];            // row scales for the two M-subtiles
  float s1 = bvals[m0 + 16 + lo];
  v2f t0 = *(const v2f*)(x3 + (m0 + lo) * 4 + 2 * hi);        // x3 tail fragments
  v2f t1 = *(const v2f*)(x3 + (m0 + 16 + lo) * 4 + 2 * hi);
#pragma unroll
  for (int nt = 0; nt < 4; ++nt) {
    int n0 = (wv * 4 + nt) * 16;
    float bias = b3[n0 + lo];
    v8f c0, c1;
#pragma unroll
    for (int v = 0; v < 8; ++v) { c0[v] = bias; c1[v] = bias; }
    for (int k0 = 0; k0 < 256; k0 += 4) {
      v2f b;
      b.x = w3[(k0 + 2 * hi + 0) * 512 + n0 + lo];
      b.y = w3[(k0 + 2 * hi + 1) * 512 + n0 + lo];
      v2f a0 = *(const v2f*)(sA + lo * 260 + k0 + 2 * hi) * s0;
      v2f a1 = *(const v2f*)(sA + (16 + lo) * 260 + k0 + 2 * hi) * s1;
      c0 = __builtin_amdgcn_wmma_f32_16x16x4_f32(false, a0, false, b, (short)0, c0, false, false);
      c1 = __builtin_amdgcn_wmma_f32_16x16x4_f32(false, a1, false, b, (short)0, c1, false, false);
    }
    { // tail K-step (K = 256..259): A = tensor_x3, unscaled
      v2f b;
      b.x = w3[(256 + 2 * hi + 0) * 512 + n0 + lo];
      b.y = w3[(256 + 2 * hi + 1) * 512 + n0 + lo];
      c0 = __builtin_amdgcn_wmma_f32_16x16x4_f32(false, t0, false, b, (short)0, c0, false, false);
      c1 = __builtin_amdgcn_wmma_f32_16x16x4_f32(false, t1, false, b, (short)0, c1, false, false);
    }
#pragma unroll
    for (int v = 0; v < 8; ++v) {
      float d0 = c0[v]; d0 = d0 > 0.f ? d0 : 0.f;
      float d1 = c1[v]; d1 = d1 > 0.f ? d1 : 0.f;
      long long r0 = m0 + v + 8 * hi;
      long long r1 = m0 + 16 + v + 8 * hi;
      h3[r0 * 512 + n0 + lo]    = d0;
      h3cat[r0 * 516 + n0 + lo] = d0;
      h3[r1 * 512 + n0 + lo]    = d1;
      h3cat[r1 * 516 + n0 + lo] = d1;
    }
  }
}

// ------------------------------------------------- h4 = h3_cat @ w4 + b4 -> h4 and logits
__global__ __launch_bounds__(256) void k_h4(const float* __restrict__ h3,
                                            const float* __restrict__ x3,
                                            const float* __restrict__ w4,
                                            const float* __restrict__ b4,
                                            float* __restrict__ h4,
                                            float* __restrict__ logits) {
  __shared__ float sW[516 * 4];
  __shared__ float sB[4];
  for (int i = threadIdx.x; i < 516 * 4; i += 256) sW[i] = w4[i];
  if (threadIdx.x < 4) sB[threadIdx.x] = b4[threadIdx.x];
  __syncthreads();
  int i = blockIdx.x * 256 + threadIdx.x;
  if (i >= N_POINTS) return;
  float a0 = sB[0], a1 = sB[1], a2 = sB[2], a3 = sB[3];
  const float* row = h3 + (long long)i * 512;
  for (int k = 0; k < 512; k += 4) {
    float4 x = *(const float4*)(row + k);
    a0 += x.x * sW[(k+0)*4+0] + x.y * sW[(k+1)*4+0] + x.z * sW[(k+2)*4+0] + x.w * sW[(k+3)*4+0];
    a1 += x.x * sW[(k+0)*4+1] + x.y * sW[(k+1)*4+1] + x.z * sW[(k+2)*4+1] + x.w * sW[(k+3)*4+1];
    a2 += x.x * sW[(k+0)*4+2] + x.y * sW[(k+1)*4+2] + x.z * sW[(k+2)*4+2] + x.w * sW[(k+3)*4+2];
    a3 += x.x * sW[(k+0)*4+3] + x.y * sW[(k+1)*4+3] + x.z * sW[(k+2)*4+3] + x.w * sW[(k+3)*4+3];
  }
#pragma unroll
  for (int k = 0; k < 4; ++k) {
    float x = x3[i * 4 + k];
    a0 += x * sW[(512+k)*4+0]; a1 += x * sW[(512+k)*4+1];
    a2 += x * sW[(512+k)*4+2]; a3 += x * sW[(512+k)*4+3];
  }
  float4 r = make_float4(a0, a1, a2, a3);
  *(float4*)(h4 + (long long)i * 4)     = r;
  *(float4*)(logits + (long long)i * 4) = r;
}

// ---------------------------------------------------------------- launcher
extern "C" void kernel_launch(void* const* d_in, const int* in_sizes, int n_in,
                              void* d_out, int out_size, void* d_ws, size_t ws_size,
                              hipStream_t stream) {
  const float* feature = (const float*)d_in[0];
  const float* x3      = (const float*)d_in[1];
  const int*   arows   = (const int*)d_in[2];
  const int*   acols   = (const int*)d_in[3];
  const float* avals   = (const float*)d_in[4];
  // d_in[5] = bing_rows (arange, unused)
  const int*   bcols   = (const int*)d_in[6];
  const float* bvals   = (const float*)d_in[7];
  const float* w1 = (const float*)d_in[8];  const float* b1 = (const float*)d_in[9];
  const float* w2 = (const float*)d_in[10]; const float* b2 = (const float*)d_in[11];
  const float* w3 = (const float*)d_in[12]; const float* b3 = (const float*)d_in[13];
  const float* w4 = (const float*)d_in[14]; const float* b4 = (const float*)d_in[15];

  float* out = (float*)d_out;
  float* o_logits = out;                               // 400000*4
  float* o_h1     = o_logits + (long long)N_POINTS*4;  // 100000*132
  float* o_h2     = o_h1 + (long long)N_POOL*132;      // 400000*512 (h3)
  float* o_h3cat  = o_h2 + (long long)N_POINTS*512;    // 400000*516
  float* o_h4     = o_h3cat + (long long)N_POINTS*516; // 400000*4

  // workspace: [xw1|agg1] reused as [xw2], then [agg2]; total 51.2M floats = 204.8 MB
  float* ws   = (float*)d_ws;
  float* xw1  = ws;
  float* agg1 = ws + (long long)N_POOL*128;
  float* xw2  = ws;                         // reuses xw1+agg1 after h1 is done
  float* agg2 = ws + (long long)N_POOL*256;

  k_zero<<<N_POOL*128/256, 256, 0, stream>>>(agg1, N_POOL*128);
  k_gemm1<<<N_POOL/16, 256, 0, stream>>>(feature, w1, xw1);
  k_spmm1<<<(long long)N_EDGES*128/256, 256, 0, stream>>>(arows, acols, avals, xw1, agg1);
  k_h1<<<(N_POOL*132 + 255)/256, 256, 0, stream>>>(agg1, b1, feature, o_h1);
  k_gemm2<<<N_POOL/32, 256, 0, stream>>>(o_h1, w2, xw2);
  k_zero<<<N_POOL*256/256, 256, 0, stream>>>(agg2, N_POOL*256);
  k_spmm2<<<N_EDGES, 256, 0, stream>>>(arows, acols, avals, xw2, agg2);
  k_h2a<<<N_POOL*256/256, 256, 0, stream>>>(agg2, b2);
  k_gemm3<<<N_POINTS/32, 256, 0, stream>>>(agg2, bcols, bvals, x3, w3, b3, o_h2, o_h3cat);
  k_h4<<<(N_POINTS + 255)/256, 256, 0, stream>>>(o_h2, x3, w4, b4, o_h4, o_logits);
}